// EnvironmentLight_45921790329432
// MI455X (gfx1250) — compile-verified
//
#include <hip/hip_runtime.h>

typedef float v2f __attribute__((ext_vector_type(2)));
typedef float v8f __attribute__((ext_vector_type(8)));

// Ramamoorthi-Hanrahan irradiance constants
#define C1d 0.429043f
#define C2d 0.511664f
#define C3d 0.743125f
#define C4d 0.886227f
#define C5d 0.247708f

__device__ __forceinline__ float sigmoidf_(float x) {
    return 1.0f / (1.0f + __expf(-x));
}

// One wave32 handles 16 pixels: pixel L lives (duplicated) in lanes L and L+16 so
// the f32 WMMA A-tile (16x4: v0 = K0|K2 across lane halves, v1 = K1|K3) needs no
// cross-lane shuffles. 256-thread block = 8 waves = 128 pixels.
__global__ __launch_bounds__(256) void EnvironmentLight_shade_kernel(
    const float* __restrict__ gb_pos,
    const float* __restrict__ gb_normal,
    const float* __restrict__ albedo,
    const float* __restrict__ kr,
    const float* __restrict__ km,
    const float* __restrict__ view_pos,
    const float* __restrict__ base,    // (25,3)
    const float* __restrict__ fg_lut,  // (256,256,2)
    float* __restrict__ out_rgb,
    float* __restrict__ out_diffuse,
    float* __restrict__ out_spec,
    int N)
{
    __shared__ float shlds[8 * 48];  // per-wave: 16 pixels x 3 channels of spec_rad

    const int tid  = threadIdx.x;
    const int lane = tid & 31;
    const int wave = tid >> 5;
    const int col  = lane & 15;         // pixel-in-wave index == B-matrix column
    const bool hi  = lane >= 16;

    const int pixBase = blockIdx.x * 128 + wave * 16;
    const int pix = pixBase + col;
    const bool valid = pix < N;
    const int p = valid ? pix : (N - 1);   // clamp loads; keep EXEC full for WMMA

    // ---------------- B-tile values (wave-invariant, L2-resident) ------------------
    // B[k][n] = base[k][n] for n<3 else 0.  Row pair per chunk c: k0 = 4c + 2*hi,
    // k1 = k0+1.  Flat index k*3+col = 12c + (6*hi + col).  Clamp col so all loads
    // stay inside base[0..74]; zero out invalid lanes with selects (no EXEC branches).
    const int  bc  = (col < 3) ? col : 0;
    const bool mc  = (col < 3);
    const float* bp = base + (hi ? 6 : 0) + bc;
    float q00 = bp[0],  q01 = bp[3];     // c=0: k = 0|2, 1|3
    float q10 = bp[12], q11 = bp[15];    // c=1
    float q20 = bp[24], q21 = bp[27];    // c=2
    float q30 = bp[36], q31 = bp[39];    // c=3
    float q40 = bp[48], q41 = bp[51];    // c=4
    float q50 = bp[60], q51 = bp[63];    // c=5
    float q60 = base[72 + bc];           // c=6: k=24 (lo half only)
    float b00 = mc ? q00 : 0.0f, b10 = mc ? q01 : 0.0f;
    float b01 = mc ? q10 : 0.0f, b11 = mc ? q11 : 0.0f;
    float b02 = mc ? q20 : 0.0f, b12 = mc ? q21 : 0.0f;
    float b03 = mc ? q30 : 0.0f, b13 = mc ? q31 : 0.0f;
    float b04 = mc ? q40 : 0.0f, b14 = mc ? q41 : 0.0f;
    float b05 = mc ? q50 : 0.0f, b15 = mc ? q51 : 0.0f;
    float b06 = (mc && !hi) ? q60 : 0.0f, b16 = 0.0f;

    // ---------------- per-pixel inputs (coalesced, dup across half-waves) ----------
    float px = gb_pos[p*3+0],    py = gb_pos[p*3+1],    pz = gb_pos[p*3+2];
    float nx = gb_normal[p*3+0], ny = gb_normal[p*3+1], nz = gb_normal[p*3+2];
    float ax = albedo[p*3+0],    ay = albedo[p*3+1],    az = albedo[p*3+2];
    float vx = view_pos[p*3+0],  vy = view_pos[p*3+1],  vz = view_pos[p*3+2];
    float r  = kr[p];
    float m  = km[p];

    // wo = normalize(view_pos - gb_pos)
    float wx = vx - px, wy = vy - py, wz = vz - pz;
    float wl = rsqrtf(fmaxf(wx*wx + wy*wy + wz*wz, 1e-20f));
    wx *= wl; wy *= wl; wz *= wl;

    float dwn = wx*nx + wy*ny + wz*nz;          // wo . n
    // reflvec = normalize(2*(wo.n)*n - wo)
    float rx = 2.0f*dwn*nx - wx;
    float ry = 2.0f*dwn*ny - wy;
    float rz = 2.0f*dwn*nz - wz;
    float rl = rsqrtf(fmaxf(rx*rx + ry*ry + rz*rz, 1e-20f));
    rx *= rl; ry *= rl; rz *= rl;

    // ---------------- diffuse irradiance: n_h^T M n_h per channel ------------------
    // base is uniform-addressed here -> scalar loads.
    float nxx = nx*nx, nyy = ny*ny, nzz = nz*nz;
    float diffuse_irr[3];
    #pragma unroll
    for (int c = 0; c < 3; ++c) {
        float e0 = base[0*3+c], e1 = base[1*3+c], e2 = base[2*3+c];
        float e3 = base[3*3+c], e4 = base[4*3+c], e5 = base[5*3+c];
        float e6 = base[6*3+c], e7 = base[7*3+c], e8 = base[8*3+c];
        diffuse_irr[c] = C1d*e8*(nxx - nyy) + C3d*e6*nzz + (C4d*e0 - C5d*e6)
                       + 2.0f*(C1d*e4*nx*ny + C1d*e7*nx*nz + C2d*e3*nx
                             + C1d*e5*ny*nz + C2d*e1*ny + C2d*e2*nz);
    }

    // ---------------- FG LUT bilinear (256x256x2), texel-center convention ---------
    float NdotV = fmaxf(dwn, 1e-4f);
    float lx = NdotV * 256.0f - 0.5f;
    float ly = r     * 256.0f - 0.5f;
    float lx0f = floorf(lx), ly0f = floorf(ly);
    float fx = lx - lx0f, fy = ly - ly0f;
    int x0 = min(max((int)lx0f, 0), 255); int x1 = min(x0 + 1, 255);
    int y0 = min(max((int)ly0f, 0), 255); int y1 = min(y0 + 1, 255);
    const float* t00 = fg_lut + (y0*256 + x0)*2;
    const float* t01 = fg_lut + (y0*256 + x1)*2;
    const float* t10 = fg_lut + (y1*256 + x0)*2;
    const float* t11 = fg_lut + (y1*256 + x1)*2;
    float fg0 = (t00[0]*(1.0f-fx) + t01[0]*fx)*(1.0f-fy) + (t10[0]*(1.0f-fx) + t11[0]*fx)*fy;
    float fg1 = (t00[1]*(1.0f-fx) + t01[1]*fx)*(1.0f-fy) + (t10[1]*(1.0f-fx) + t11[1]*fx)*fy;

    // ---------------- SH basis (deg 0..4) on reflvec, scaled by GW kernel ----------
    // Named scalars (NOT an array) so the A-tile selects below stay single cndmasks.
    float xx = rx*rx, yy = ry*ry, zz = rz*rz;
    float xy = rx*ry, yz = ry*rz, xz = rx*rz;
    float t  = r*r;
    float g1 = __expf(-2.0f*t);      // exp(-l(l+1)kr^2), l=1
    float g2 = g1*g1*g1;             // l=2: exp(-6t)
    float g3 = g2*g2;                // l=3: exp(-12t)
    float g4 = g3*g2*g1;             // l=4: exp(-20t)

    float s0  = 0.28209479177387814f;
    float s1  = -0.4886025119029199f * ry * g1;
    float s2  =  0.4886025119029199f * rz * g1;
    float s3  = -0.4886025119029199f * rx * g1;
    float s4  =  1.0925484305920792f * xy * g2;
    float s5  = -1.0925484305920792f * yz * g2;
    float s6  =  0.31539156525252005f * (2.0f*zz - xx - yy) * g2;
    float s7  = -1.0925484305920792f * xz * g2;
    float s8  =  0.5462742152960396f * (xx - yy) * g2;
    float s9  = -0.5900435899266435f * ry * (3.0f*xx - yy) * g3;
    float s10 =  2.890611442640554f  * xy * rz * g3;
    float s11 = -0.4570457994644658f * ry * (4.0f*zz - xx - yy) * g3;
    float s12 =  0.3731763325901154f * rz * (2.0f*zz - 3.0f*xx - 3.0f*yy) * g3;
    float s13 = -0.4570457994644658f * rx * (4.0f*zz - xx - yy) * g3;
    float s14 =  1.445305721320277f  * rz * (xx - yy) * g3;
    float s15 = -0.5900435899266435f * rx * (xx - 3.0f*yy) * g3;
    float s16 =  2.5033429417967046f * xy * (xx - yy) * g4;
    float s17 = -1.7701307697799304f * yz * (3.0f*xx - yy) * g4;
    float s18 =  0.9461746957575601f * xy * (7.0f*zz - 1.0f) * g4;
    float s19 = -0.6690465435572892f * yz * (7.0f*zz - 3.0f) * g4;
    float s20 =  0.10578554691520431f * (zz*(35.0f*zz - 30.0f) + 3.0f) * g4;
    float s21 = -0.6690465435572892f * xz * (7.0f*zz - 3.0f) * g4;
    float s22 =  0.47308734787878004f * (xx - yy) * (7.0f*zz - 1.0f) * g4;
    float s23 = -1.7701307697799304f * xz * (xx - yy) * g4;
    float s24 =  0.6258357354491761f * (xx*(xx - 3.0f*yy) - yy*(3.0f*xx - yy)) * g4;

    // ---------------- spec_rad = shg(16x28) @ base(28x3-pad-16) via f32 WMMA -------
    v8f acc = {};
#define WMMA_CHUNK(lo0, lo1, hi0, hi1, Bx, By)                                   \
    do {                                                                         \
        v2f A_; A_.x = hi ? (hi0) : (lo0); A_.y = hi ? (hi1) : (lo1);            \
        v2f B_; B_.x = (Bx); B_.y = (By);                                        \
        acc = __builtin_amdgcn_wmma_f32_16x16x4_f32(                             \
            false, A_, false, B_, (short)0, acc, false, false);                  \
    } while (0)

    WMMA_CHUNK(s0,  s1,  s2,  s3,  b00, b10);
    WMMA_CHUNK(s4,  s5,  s6,  s7,  b01, b11);
    WMMA_CHUNK(s8,  s9,  s10, s11, b02, b12);
    WMMA_CHUNK(s12, s13, s14, s15, b03, b13);
    WMMA_CHUNK(s16, s17, s18, s19, b04, b14);
    WMMA_CHUNK(s20, s21, s22, s23, b05, b15);
    WMMA_CHUNK(s24, 0.0f, 0.0f, 0.0f, b06, b16);
#undef WMMA_CHUNK

    // D layout: VGPR r holds M=r (lanes 0-15) / M=r+8 (lanes 16-31), N=lane%16.
    // Scatter the 3 useful columns to per-pixel slots through LDS.
    if (col < 3) {
        #pragma unroll
        for (int rr = 0; rr < 8; ++rr) {
            shlds[wave*48 + (rr + (hi ? 8 : 0))*3 + col] = acc[rr];
        }
    }
    __syncthreads();

    float sr0 = shlds[wave*48 + col*3 + 0];
    float sr1 = shlds[wave*48 + col*3 + 1];
    float sr2 = shlds[wave*48 + col*3 + 2];

    // ---------------- combine ------------------------------------------------------
    float km1 = 1.0f - m;
    float F0x = km1*0.04f + ax*m;
    float F0y = km1*0.04f + ay*m;
    float F0z = km1*0.04f + az*m;

    float spec0 = sr0 * (F0x*fg0 + fg1);
    float spec1 = sr1 * (F0y*fg0 + fg1);
    float spec2 = sr2 * (F0z*fg0 + fg1);

    float dif0 = ax * diffuse_irr[0];
    float dif1 = ay * diffuse_irr[1];
    float dif2 = az * diffuse_irr[2];

    if (!hi && valid) {
        const int o = pix * 3;
        out_rgb[o+0] = sigmoidf_(dif0 + spec0);
        out_rgb[o+1] = sigmoidf_(dif1 + spec1);
        out_rgb[o+2] = sigmoidf_(dif2 + spec2);
        out_diffuse[o+0] = dif0;
        out_diffuse[o+1] = dif1;
        out_diffuse[o+2] = dif2;
        out_spec[o+0] = spec0;
        out_spec[o+1] = spec1;
        out_spec[o+2] = spec2;
    }
}

extern "C" void kernel_launch(void* const* d_in, const int* in_sizes, int n_in,
                              void* d_out, int out_size, void* d_ws, size_t ws_size,
                              hipStream_t stream) {
    const float* gb_pos    = (const float*)d_in[0];
    const float* gb_normal = (const float*)d_in[1];
    const float* albedo    = (const float*)d_in[2];
    // d_in[3] = ks: unused by the reference shade()
    const float* kr        = (const float*)d_in[4];
    const float* km        = (const float*)d_in[5];
    const float* view_pos  = (const float*)d_in[6];
    const float* base      = (const float*)d_in[7];
    const float* fg_lut    = (const float*)d_in[8];

    const int N = in_sizes[0] / 3;           // gb_pos is (N,3)
    float* out = (float*)d_out;              // [rgb | diffuse | spec], each (N,3)
    float* out_rgb     = out;
    float* out_diffuse = out + (size_t)3 * N;
    float* out_spec    = out + (size_t)6 * N;

    const int blocks = (N + 127) / 128;      // 128 pixels per 256-thread block
    EnvironmentLight_shade_kernel<<<blocks, 256, 0, stream>>>(
        gb_pos, gb_normal, albedo, kr, km, view_pos, base, fg_lut,
        out_rgb, out_diffuse, out_spec, N);
}